// StickyNNModel_28363964023162
// MI455X (gfx1250) — compile-verified
//
#include <hip/hip_runtime.h>
#include <hip/hip_bf16.h>

typedef __attribute__((ext_vector_type(2))) float v2f;
typedef __attribute__((ext_vector_type(8))) float v8f;

#define VOCAB   65
#define EMBED   6
#define CPS     8          // padded class_prob row stride (floats)
#define BSZ     4
#define TSZ     8192
#define SPLIT   4          // s-dimension split per row-tile

// ---------------------------------------------------------------------------
// Kernel 1: class_prob = softmax(embed_w[x]) stored padded [B*T, 8]; zero score
// ---------------------------------------------------------------------------
__global__ void k_classprob(const int* __restrict__ x,
                            const float* __restrict__ embed_w,
                            float* __restrict__ cp,
                            float* __restrict__ score) {
    int i = blockIdx.x * blockDim.x + threadIdx.x;
    if (i >= BSZ * TSZ) return;
    score[i] = 0.0f;
    int tok = x[i];
    float e[EMBED];
    float m = -1e30f;
#pragma unroll
    for (int j = 0; j < EMBED; ++j) {
        e[j] = embed_w[tok * EMBED + j];
        m = fmaxf(m, e[j]);
    }
    float s = 0.0f;
#pragma unroll
    for (int j = 0; j < EMBED; ++j) { e[j] = __expf(e[j] - m); s += e[j]; }
    float inv = 1.0f / s;
    float4 lo = make_float4(e[0]*inv, e[1]*inv, e[2]*inv, e[3]*inv);
    float4 hi = make_float4(e[4]*inv, e[5]*inv, 0.0f, 0.0f);
    float4* dst = (float4*)(cp + (size_t)i * CPS);
    dst[0] = lo;
    dst[1] = hi;
}

// ---------------------------------------------------------------------------
// Kernel 2: score[b,t] = sum_s (cp[t]·cp[s]) * mask[t,s]   (mask pre-tril'd)
// One wave per (row-tile r, s-chunk). sim tile via V_WMMA_F32_16X16X4_F32.
// Mask tile loaded ONCE (non-temporal) and reused across all 4 batches.
// ---------------------------------------------------------------------------
__global__ void __launch_bounds__(32)
k_score(const float* __restrict__ cp,
        const float* __restrict__ pm,
        float* __restrict__ score) {
    const int lane  = threadIdx.x;
    const int r     = (int)(gridDim.x - 1) - (int)blockIdx.x;  // largest-first
    const int chunk = blockIdx.y;
    const int T0    = r * 16;
    const int mrow  = lane & 15;
    const int kb    = (lane >> 4) * 2;      // K pair base: lanes 0-15 -> 0, 16-31 -> 2
    const int hhi   = (lane >> 4) * 8;      // C-layout row offset for upper half-wave

    // A fragments (rows t of this tile), one pair per batch, constant over s loop.
    v2f a0[BSZ], a1[BSZ];
#pragma unroll
    for (int b = 0; b < BSZ; ++b) {
        const float* base = cp + ((size_t)(b * TSZ + T0 + mrow)) * CPS;
        a0[b] = *(const v2f*)(base + kb);        // K = kb, kb+1
        a1[b] = *(const v2f*)(base + kb + 4);    // K = kb+4, kb+5  (6,7 are zero pad)
    }

    float sacc[BSZ][8];
#pragma unroll
    for (int b = 0; b < BSZ; ++b)
#pragma unroll
        for (int j = 0; j < 8; ++j) sacc[b][j] = 0.0f;

    for (int st = chunk; st <= r; st += SPLIT) {
        const int S0 = st * 16;

        // Mask tile, in C/D register layout: vgpr j <-> row T0+j+hhi, lane&15 <-> col
        float w[8];
        {
            const float* prow = pm + (size_t)(T0 + hhi) * TSZ + S0 + mrow;
#pragma unroll
            for (int j = 0; j < 8; ++j)
                w[j] = __builtin_nontemporal_load(prow + (size_t)j * TSZ);
        }

#pragma unroll
        for (int b = 0; b < BSZ; ++b) {
            const float* sbase = cp + ((size_t)(b * TSZ + S0 + mrow)) * CPS;
            v2f b0 = *(const v2f*)(sbase + kb);      // B[k][n] = cp[S0+n][k]
            v2f b1 = *(const v2f*)(sbase + kb + 4);
            v8f c = {};
            c = __builtin_amdgcn_wmma_f32_16x16x4_f32(false, a0[b], false, b0,
                                                      (short)0, c, false, false);
            c = __builtin_amdgcn_wmma_f32_16x16x4_f32(false, a1[b], false, b1,
                                                      (short)0, c, false, false);
#pragma unroll
            for (int j = 0; j < 8; ++j)
                sacc[b][j] = fmaf(c[j], w[j], sacc[b][j]);
        }
    }

    // Reduce across the 16 column-lanes of each half-wave, then atomicAdd.
#pragma unroll
    for (int b = 0; b < BSZ; ++b) {
#pragma unroll
        for (int j = 0; j < 8; ++j) {
            float v = sacc[b][j];
#pragma unroll
            for (int off = 1; off < 16; off <<= 1)
                v += __shfl_xor(v, off, 32);
            if ((lane & 15) == 0) {
                int row = T0 + j + hhi;           // lane 0 -> row j, lane 16 -> row j+8
                atomicAdd(&score[b * TSZ + row], v);
            }
        }
    }
}

// ---------------------------------------------------------------------------
// Kernel 3: gate + 6x6 transitions + 6x65 projection. One wave per token.
// ---------------------------------------------------------------------------
__global__ void k_output(const float* __restrict__ cp,
                         const float* __restrict__ score,
                         const float* __restrict__ reset_threshold,
                         const float* __restrict__ temperature,
                         const float* __restrict__ Ct,   // class_transition [6,6]
                         const float* __restrict__ Rt,   // reset_transition [6,6]
                         const float* __restrict__ Wout, // output [6,65]
                         float* __restrict__ out) {
    int gtid = blockIdx.x * blockDim.x + threadIdx.x;
    int tokn = gtid >> 5;            // one wave per token
    int lane = gtid & 31;
    if (tokn >= BSZ * TSZ) return;

    const float* c = cp + (size_t)tokn * CPS;
    float cpv[EMBED];
#pragma unroll
    for (int j = 0; j < EMBED; ++j) cpv[j] = c[j];

    float th = reset_threshold[0];
    float tm = temperature[0];
    float z  = (score[tokn] - th * th) / (tm * tm);
    float rr = 1.0f / (1.0f + __expf(-z));

    float nc[EMBED];
#pragma unroll
    for (int e = 0; e < EMBED; ++e) {
        float tc = 0.0f, rc = 0.0f;
#pragma unroll
        for (int f = 0; f < EMBED; ++f) {
            tc = fmaf(cpv[f], Ct[f * EMBED + e], tc);
            rc = fmaf(cpv[f], Rt[f * EMBED + e], rc);
        }
        nc[e] = rr * rc + (1.0f - rr) * tc;
    }

    for (int v = lane; v < VOCAB; v += 32) {
        float o = 0.0f;
#pragma unroll
        for (int e = 0; e < EMBED; ++e)
            o = fmaf(nc[e], Wout[e * VOCAB + v], o);
        out[(size_t)tokn * VOCAB + v] = o;
    }
}

// ---------------------------------------------------------------------------
extern "C" void kernel_launch(void* const* d_in, const int* in_sizes, int n_in,
                              void* d_out, int out_size, void* d_ws, size_t ws_size,
                              hipStream_t stream) {
    const int*   x     = (const int*)  d_in[0];
    const float* emb_w = (const float*)d_in[1];
    const float* pm    = (const float*)d_in[2];
    const float* rth   = (const float*)d_in[3];
    const float* temp  = (const float*)d_in[4];
    const float* Ct    = (const float*)d_in[5];
    const float* Rt    = (const float*)d_in[6];
    const float* Wout  = (const float*)d_in[7];
    float*       out   = (float*)d_out;

    // Workspace layout: cp [B*T*8] floats, then score [B*T] floats
    float* cp    = (float*)d_ws;
    float* score = cp + (size_t)BSZ * TSZ * CPS;

    const int ntok = BSZ * TSZ;

    k_classprob<<<(ntok + 255) / 256, 256, 0, stream>>>(x, emb_w, cp, score);

    dim3 grid2(TSZ / 16, SPLIT);          // 512 row-tiles x 4 s-chunks
    k_score<<<grid2, 32, 0, stream>>>(cp, pm, score);

    // one wave (32 lanes) per token, 8 tokens per 256-thread block
    k_output<<<(ntok * 32 + 255) / 256, 256, 0, stream>>>(
        cp, score, rth, temp, Ct, Rt, Wout, out);
}